// GFCN_7395933683880
// MI455X (gfx1250) — compile-verified
//
#include <hip/hip_runtime.h>
#include <cstdint>
#include <cstddef>

// ---------------------------------------------------------------------------
// GFCN (graclus U-Net with SplineConv) forward for gfx1250 (MI455X, wave32).
// GEMMs use V_WMMA_F32_16X16X4_F32 (full fp32 -> matches f32 reference).
// Each wave computes ALL 16-col tiles of a 16-row strip so the big scattered
// S matrix (the dominant HBM traffic) is streamed exactly once per layer.
// ---------------------------------------------------------------------------

typedef float v2f __attribute__((ext_vector_type(2)));
typedef float v8f __attribute__((ext_vector_type(8)));

#define TPB 256

__device__ __forceinline__ unsigned fenc(float f) {
    unsigned u = __float_as_uint(f);
    return (u & 0x80000000u) ? ~u : (u | 0x80000000u);
}
__device__ __forceinline__ float fdec(unsigned u) {
    unsigned b = (u & 0x80000000u) ? (u & 0x7FFFFFFFu) : ~u;
    return __uint_as_float(b);
}
__device__ __forceinline__ float eluf(float v) { return v > 0.f ? v : expm1f(v); }

// ------------------------- prep kernels ------------------------------------

__global__ void k_deg(const int* __restrict__ dst, float* deg, int E) {
    int e = blockIdx.x * blockDim.x + threadIdx.x;
    if (e < E) atomicAdd(&deg[dst[e]], 1.0f);
}

// global max |cart| over all edges & both components (values >= 0 so raw bits
// are order-preserving as unsigned). Wave-reduce first to cut atomics.
__global__ void k_maxabs(const float* __restrict__ pos, const int* __restrict__ src,
                         const int* __restrict__ dst, unsigned* mx, int E) {
    int e = blockIdx.x * blockDim.x + threadIdx.x;
    float m = 0.f;
    if (e < E) {
        int s = src[e], d = dst[e];
        float cx = fabsf(pos[2 * d + 0] - pos[2 * s + 0]);
        float cy = fabsf(pos[2 * d + 1] - pos[2 * s + 1]);
        m = fmaxf(cx, cy);
    }
    for (int off = 16; off; off >>= 1)
        m = fmaxf(m, __shfl_xor(m, off, 32));
    if ((threadIdx.x & 31) == 0)
        atomicMax(mx, __float_as_uint(m));
}

// Per-edge: 4 bilinear-spline corners. weight is pre-divided by max(deg[dst],1)
// so the whole layer becomes  out = S @ W + x @ root + b.
__global__ void k_corners(const float* __restrict__ pos, const int* __restrict__ src,
                          const int* __restrict__ dst, const unsigned* __restrict__ mx,
                          const float* __restrict__ deg,
                          int4* __restrict__ ek, float4* __restrict__ ew, int E) {
    int e = blockIdx.x * blockDim.x + threadIdx.x;
    if (e >= E) return;
    float m = __uint_as_float(*mx);
    int s = src[e], d = dst[e];
    float inv = 0.5f / m;
    float e0 = (pos[2 * d + 0] - pos[2 * s + 0]) * inv + 0.5f;
    float e1 = (pos[2 * d + 1] - pos[2 * s + 1]) * inv + 0.5f;
    float v0 = e0 * 4.f, v1 = e1 * 4.f;
    float l0 = floorf(v0), l1 = floorf(v1);
    float f0 = v0 - l0, f1 = v1 - l1;
    int i0 = (int)l0, i1 = (int)l1;
    int i00 = min(max(i0, 0), 4), i01 = min(max(i0 + 1, 0), 4);
    int i10 = min(max(i1, 0), 4), i11 = min(max(i1 + 1, 0), 4);
    float idg = 1.f / fmaxf(deg[d], 1.f);
    int4 k; float4 w;
    k.x = i00 + i10 * 5; w.x = (1.f - f0) * (1.f - f1) * idg;
    k.y = i01 + i10 * 5; w.y = f0 * (1.f - f1) * idg;
    k.z = i00 + i11 * 5; w.z = (1.f - f0) * f1 * idg;
    k.w = i01 + i11 * 5; w.w = f0 * f1 * idg;
    ek[e] = k; ew[e] = w;
}

__global__ void k_cnt(const int* __restrict__ cl, float* cnt, int n) {
    int i = blockIdx.x * blockDim.x + threadIdx.x;
    if (i < n) atomicAdd(&cnt[cl[i]], 1.0f);
}

__global__ void k_possum(const float* __restrict__ pos, const int* __restrict__ cl,
                         float* pos1, int n) {
    int i = blockIdx.x * blockDim.x + threadIdx.x;
    if (i >= n) return;
    int c = cl[i];
    atomicAdd(&pos1[2 * c + 0], pos[2 * i + 0]);
    atomicAdd(&pos1[2 * c + 1], pos[2 * i + 1]);
}

__global__ void k_posdiv(float* pos1, const float* __restrict__ cnt, int n1) {
    int j = blockIdx.x * blockDim.x + threadIdx.x;
    if (j >= n1) return;
    float inv = 1.f / fmaxf(cnt[j], 1.f);
    pos1[2 * j + 0] *= inv;
    pos1[2 * j + 1] *= inv;
}

// ------------------------- spline scatter ----------------------------------

__global__ void k_scatter(const float* __restrict__ x, const int* __restrict__ src,
                          const int* __restrict__ dst,
                          const int4* __restrict__ ek, const float4* __restrict__ ew,
                          float* S, int cin, int stride, int E) {
    size_t tid = (size_t)blockIdx.x * blockDim.x + threadIdx.x;
    size_t total = (size_t)E * cin;
    if (tid >= total) return;
    int e = (int)(tid / cin);
    int i = (int)(tid - (size_t)e * cin);
    float xv = x[(size_t)src[e] * cin + i];
    int4 k = ek[e]; float4 w = ew[e];
    float* row = S + (size_t)dst[e] * stride + i;
    atomicAdd(row + k.x * cin, w.x * xv);
    atomicAdd(row + k.y * cin, w.y * xv);
    atomicAdd(row + k.z * cin, w.z * xv);
    atomicAdd(row + k.w * cin, w.w * xv);
}

__global__ void k_append(const float* __restrict__ x, float* S, int n, int cin,
                         int Ks, int stride) {
    size_t tid = (size_t)blockIdx.x * blockDim.x + threadIdx.x;
    if (tid >= (size_t)n * cin) return;
    int j = (int)(tid / cin);
    int i = (int)(tid - (size_t)j * cin);
    S[(size_t)j * stride + Ks + i] = x[tid];
}

__global__ void k_packw(const float* __restrict__ W, const float* __restrict__ root,
                        float* Wext, int Ks, int cin, int cout, int Ke) {
    size_t tid = (size_t)blockIdx.x * blockDim.x + threadIdx.x;
    if (tid >= (size_t)Ke * cout) return;
    int k = (int)(tid / cout);
    int o = (int)(tid - (size_t)k * cout);
    float v = 0.f;
    if (k < Ks)             v = W[(size_t)k * cout + o];
    else if (k < Ks + cin)  v = root[(size_t)(k - Ks) * cout + o];
    Wext[tid] = v;
}

// ------------------------- WMMA fp32 GEMM + bias + ELU ---------------------
// out[n, cout] = elu( Sext[n, K] @ Wext[K, cout] + bias[cout] )
// One wave per 16-row strip; it holds NT = cout/16 accumulators and reuses the
// A fragment across all N-tiles => S is streamed from HBM exactly once.
// A 16x4 layout: lanes 0-15 hold K={0,1}, lanes 16-31 hold K={2,3}.
// C/D layout: vgpr r -> row r (lanes 0-15) / row r+8 (lanes 16-31), col=lane&15.
template <int NT>
__global__ void k_wmma_gemm(const float* __restrict__ S, const float* __restrict__ W,
                            const float* __restrict__ bias, float* __restrict__ out,
                            int n, int K, int cout) {
    int lane  = threadIdx.x;                               // 0..31 (wave32)
    int mtile = blockIdx.x * blockDim.y + threadIdx.y;
    int row0  = mtile * 16;
    if (row0 >= n) return;                                 // wave-uniform exit

    int m      = lane & 15;
    int khalf  = (lane >> 4) << 1;                         // 0 or 2
    int row    = row0 + m;
    bool rvld  = row < n;
    const float* srow = S + (size_t)(rvld ? row : 0) * K;
    int nsub   = lane & 15;

    v8f acc[NT];
    const v8f zero = {0.f, 0.f, 0.f, 0.f, 0.f, 0.f, 0.f, 0.f};
#pragma unroll
    for (int t = 0; t < NT; ++t) acc[t] = zero;

    for (int k = 0; k < K; k += 4) {
        float a0 = srow[k + khalf];
        float a1 = srow[k + khalf + 1];
        v2f a;
        a.x = rvld ? a0 : 0.f;
        a.y = rvld ? a1 : 0.f;
        const float* wr0 = W + (size_t)(k + khalf) * cout + nsub;
        const float* wr1 = W + (size_t)(k + khalf + 1) * cout + nsub;
#pragma unroll
        for (int t = 0; t < NT; ++t) {
            v2f b;
            b.x = wr0[t * 16];
            b.y = wr1[t * 16];
            acc[t] = __builtin_amdgcn_wmma_f32_16x16x4_f32(false, a, false, b,
                                                           (short)0, acc[t],
                                                           false, false);
        }
    }

    int rbase = row0 + ((lane >> 4) << 3);
#pragma unroll
    for (int t = 0; t < NT; ++t) {
        int col = t * 16 + nsub;
        float bval = bias[col];
#pragma unroll
        for (int r = 0; r < 8; ++r) {
            int rr = rbase + r;
            if (rr < n) out[(size_t)rr * cout + col] = eluf(acc[t][r] + bval);
        }
    }
}

// ------------------------- pooling / bweights ------------------------------

__global__ void k_centsum(const float* __restrict__ x, const int* __restrict__ cl,
                          float* cent, int n, int C) {
    size_t tid = (size_t)blockIdx.x * blockDim.x + threadIdx.x;
    if (tid >= (size_t)n * C) return;
    int j = (int)(tid / C);
    int ch = (int)(tid - (size_t)j * C);
    atomicAdd(&cent[(size_t)cl[j] * C + ch], x[tid]);
}

__global__ void k_wdiv(const float* __restrict__ x, const int* __restrict__ cl,
                       const float* __restrict__ cent, const float* __restrict__ cnt,
                       float* w, int n, int C) {
    size_t tid = (size_t)blockIdx.x * blockDim.x + threadIdx.x;
    if (tid >= (size_t)n * C) return;
    int j = (int)(tid / C);
    int ch = (int)(tid - (size_t)j * C);
    int c = cl[j];
    float mean = cent[(size_t)c * C + ch] / fmaxf(cnt[c], 1.f);
    float r = x[tid] / mean;
    if (isnan(r)) r = 0.f;
    w[tid] = r;
}

__global__ void k_initmax(unsigned* bits, size_t n) {
    size_t tid = (size_t)blockIdx.x * blockDim.x + threadIdx.x;
    if (tid < n) bits[tid] = 0x007FFFFFu;   // fenc(-inf)
}

__global__ void k_segmax(const float* __restrict__ x, const int* __restrict__ cl,
                         unsigned* bits, int n, int C) {
    size_t tid = (size_t)blockIdx.x * blockDim.x + threadIdx.x;
    if (tid >= (size_t)n * C) return;
    int j = (int)(tid / C);
    int ch = (int)(tid - (size_t)j * C);
    atomicMax(&bits[(size_t)cl[j] * C + ch], fenc(x[tid]));
}

__global__ void k_decode(const unsigned* __restrict__ bits, float* x, size_t n) {
    size_t tid = (size_t)blockIdx.x * blockDim.x + threadIdx.x;
    if (tid < n) x[tid] = fdec(bits[tid]);
}

// out[j,ch] = w[j,ch] * up[cl[j], ch]
__global__ void k_unpool(const float* __restrict__ w, const float* __restrict__ up,
                         const int* __restrict__ cl, float* out, int n, int C) {
    size_t tid = (size_t)blockIdx.x * blockDim.x + threadIdx.x;
    if (tid >= (size_t)n * C) return;
    int j = (int)(tid / C);
    int ch = (int)(tid - (size_t)j * C);
    out[tid] = w[tid] * up[(size_t)cl[j] * C + ch];
}

// ------------------------- layer 4: dot (cout == 1) + elu + sigmoid --------
// One wave per node; float4 strided loads (K is a multiple of 4, rows 16B
// aligned since stride*4 bytes is a multiple of 16 and S is 256B aligned).

__global__ void k_dot_sig(const float* __restrict__ S, const float* __restrict__ w,
                          const float* __restrict__ b4, float* __restrict__ out,
                          int n, int K) {
    int lane = threadIdx.x;
    int node = blockIdx.x * blockDim.y + threadIdx.y;
    if (node >= n) return;
    const float* srow = S + (size_t)node * K;
    float acc = 0.f;
    for (int k = lane * 4; k < K; k += 128) {
        float4 sv = *(const float4*)(srow + k);
        float4 wv = *(const float4*)(w + k);
        acc += sv.x * wv.x + sv.y * wv.y + sv.z * wv.z + sv.w * wv.w;
    }
    for (int off = 16; off; off >>= 1) acc += __shfl_down(acc, off, 32);
    if (lane == 0) {
        float v = eluf(acc + b4[0]);
        out[node] = 1.f / (1.f + expf(-v));
    }
}

// ---------------------------------------------------------------------------

static inline int blks(size_t n) { return (int)((n + TPB - 1) / TPB); }

extern "C" void kernel_launch(void* const* d_in, const int* in_sizes, int n_in,
                              void* d_out, int out_size, void* d_ws, size_t ws_size,
                              hipStream_t stream) {
    (void)n_in; (void)ws_size; (void)out_size;

    const float* x     = (const float*)d_in[0];
    const float* pos0  = (const float*)d_in[1];
    const float* W1    = (const float*)d_in[2];
    const float* root1 = (const float*)d_in[3];
    const float* b1    = (const float*)d_in[4];
    const float* W2    = (const float*)d_in[5];
    const float* root2 = (const float*)d_in[6];
    const float* b2    = (const float*)d_in[7];
    const float* W3    = (const float*)d_in[8];
    const float* root3 = (const float*)d_in[9];
    const float* b3    = (const float*)d_in[10];
    const float* W4    = (const float*)d_in[11];
    const float* root4 = (const float*)d_in[12];
    const float* b4    = (const float*)d_in[13];
    const int*   ei0   = (const int*)d_in[14];
    const int*   ei1   = (const int*)d_in[15];
    const int*   cl1   = (const int*)d_in[16];
    const int*   cl2   = (const int*)d_in[17];
    float* out = (float*)d_out;

    const int n0 = in_sizes[0];
    const int E0 = in_sizes[14] / 2;
    const int E1 = in_sizes[15] / 2;
    const int n1 = in_sizes[17];

    const int* src0 = ei0;        const int* dst0 = ei0 + E0;
    const int* src1 = ei1;        const int* dst1 = ei1 + E1;

    // ---- workspace carve-out --------------------------------------------
    size_t off = 0;
    auto alloc = [&](size_t bytes) -> void* {
        void* p = (char*)d_ws + off;
        off += (bytes + 255) & ~(size_t)255;
        return p;
    };
    size_t szS = (size_t)n0 * 832;                       // layers 2/4
    size_t s3  = (size_t)n1 * 1664;                      // layer 3
    if (s3 > szS) szS = s3;

    float*    S      = (float*)alloc(szS * 4);
    int4*     ed0k   = (int4*)alloc((size_t)E0 * 16);
    float4*   ed0w   = (float4*)alloc((size_t)E0 * 16);
    int4*     ed1k   = (int4*)alloc((size_t)E1 * 16);
    float4*   ed1w   = (float4*)alloc((size_t)E1 * 16);
    float*    deg0   = (float*)alloc((size_t)n0 * 4);
    float*    deg1   = (float*)alloc((size_t)n1 * 4);
    float*    pos1   = (float*)alloc((size_t)n1 * 8);
    float*    cnt1   = (float*)alloc((size_t)n1 * 4);
    float*    cnt2   = (float*)alloc((size_t)n1 * 4);
    unsigned* mx     = (unsigned*)alloc(256);            // [0]=level0, [1]=level1
    float*    xa     = (float*)alloc((size_t)n0 * 32 * 4);
    float*    w1     = (float*)alloc((size_t)n0 * 32 * 4);
    unsigned* x1bits = (unsigned*)alloc((size_t)n1 * 32 * 4);
    float*    x1     = (float*)alloc((size_t)n1 * 32 * 4);
    float*    xb     = (float*)alloc((size_t)n1 * 64 * 4);
    float*    cent   = (float*)alloc((size_t)n1 * 64 * 4);
    float*    w2     = (float*)alloc((size_t)n1 * 64 * 4);
    unsigned* x2bits = (unsigned*)alloc((size_t)n1 * 64 * 4);
    float*    x2     = (float*)alloc((size_t)n1 * 64 * 4);
    float*    xu     = (float*)alloc((size_t)n1 * 64 * 4);
    float*    xc     = (float*)alloc((size_t)n1 * 32 * 4);
    float*    x0u    = (float*)alloc((size_t)n0 * 32 * 4);
    float*    Wext   = (float*)alloc((size_t)1664 * 64 * 4);

    // ---- level-0 prep ----------------------------------------------------
    hipMemsetAsync(mx, 0, 8, stream);
    hipMemsetAsync(deg0, 0, (size_t)n0 * 4, stream);
    k_deg<<<blks(E0), TPB, 0, stream>>>(dst0, deg0, E0);
    k_maxabs<<<blks(E0), TPB, 0, stream>>>(pos0, src0, dst0, &mx[0], E0);
    k_corners<<<blks(E0), TPB, 0, stream>>>(pos0, src0, dst0, &mx[0], deg0, ed0k, ed0w, E0);

    // ---- level-1 prep (pos1 = seg_mean(pos0, cluster1)) ------------------
    hipMemsetAsync(pos1, 0, (size_t)n1 * 8, stream);
    hipMemsetAsync(cnt1, 0, (size_t)n1 * 4, stream);
    k_cnt<<<blks(n0), TPB, 0, stream>>>(cl1, cnt1, n0);
    k_possum<<<blks(n0), TPB, 0, stream>>>(pos0, cl1, pos1, n0);
    k_posdiv<<<blks(n1), TPB, 0, stream>>>(pos1, cnt1, n1);
    hipMemsetAsync(deg1, 0, (size_t)n1 * 4, stream);
    k_deg<<<blks(E1), TPB, 0, stream>>>(dst1, deg1, E1);
    k_maxabs<<<blks(E1), TPB, 0, stream>>>(pos1, src1, dst1, &mx[1], E1);
    k_corners<<<blks(E1), TPB, 0, stream>>>(pos1, src1, dst1, &mx[1], deg1, ed1k, ed1w, E1);

    dim3 gblk(32, 4);

    // ---- layer 1: cin=1, cout=32, Ks=25, Ke=28 ---------------------------
    hipMemsetAsync(S, 0, (size_t)n0 * 28 * 4, stream);
    k_scatter<<<blks((size_t)E0), TPB, 0, stream>>>(x, src0, dst0, ed0k, ed0w, S, 1, 28, E0);
    k_append<<<blks((size_t)n0), TPB, 0, stream>>>(x, S, n0, 1, 25, 28);
    k_packw<<<blks((size_t)28 * 32), TPB, 0, stream>>>(W1, root1, Wext, 25, 1, 32, 28);
    {
        int mt = (n0 + 15) / 16;
        dim3 grid((mt + 3) / 4);
        k_wmma_gemm<2><<<grid, gblk, 0, stream>>>(S, Wext, b1, xa, n0, 28, 32);
    }

    // ---- bweights(w1) + pool(x1) ----------------------------------------
    hipMemsetAsync(cent, 0, (size_t)n1 * 32 * 4, stream);
    k_centsum<<<blks((size_t)n0 * 32), TPB, 0, stream>>>(xa, cl1, cent, n0, 32);
    k_wdiv<<<blks((size_t)n0 * 32), TPB, 0, stream>>>(xa, cl1, cent, cnt1, w1, n0, 32);
    k_initmax<<<blks((size_t)n1 * 32), TPB, 0, stream>>>(x1bits, (size_t)n1 * 32);
    k_segmax<<<blks((size_t)n0 * 32), TPB, 0, stream>>>(xa, cl1, x1bits, n0, 32);
    k_decode<<<blks((size_t)n1 * 32), TPB, 0, stream>>>(x1bits, x1, (size_t)n1 * 32);

    // ---- layer 2: cin=32, cout=64, Ks=800, Ke=832 ------------------------
    hipMemsetAsync(S, 0, (size_t)n1 * 832 * 4, stream);
    k_scatter<<<blks((size_t)E1 * 32), TPB, 0, stream>>>(x1, src1, dst1, ed1k, ed1w, S, 32, 832, E1);
    k_append<<<blks((size_t)n1 * 32), TPB, 0, stream>>>(x1, S, n1, 32, 800, 832);
    k_packw<<<blks((size_t)832 * 64), TPB, 0, stream>>>(W2, root2, Wext, 800, 32, 64, 832);
    {
        int mt = (n1 + 15) / 16;
        dim3 grid((mt + 3) / 4);
        k_wmma_gemm<4><<<grid, gblk, 0, stream>>>(S, Wext, b2, xb, n1, 832, 64);
    }

    // ---- bweights(w2) + pool(x2) + unpool(xu) ----------------------------
    hipMemsetAsync(cnt2, 0, (size_t)n1 * 4, stream);
    hipMemsetAsync(cent, 0, (size_t)n1 * 64 * 4, stream);
    k_cnt<<<blks(n1), TPB, 0, stream>>>(cl2, cnt2, n1);
    k_centsum<<<blks((size_t)n1 * 64), TPB, 0, stream>>>(xb, cl2, cent, n1, 64);
    k_wdiv<<<blks((size_t)n1 * 64), TPB, 0, stream>>>(xb, cl2, cent, cnt2, w2, n1, 64);
    k_initmax<<<blks((size_t)n1 * 64), TPB, 0, stream>>>(x2bits, (size_t)n1 * 64);
    k_segmax<<<blks((size_t)n1 * 64), TPB, 0, stream>>>(xb, cl2, x2bits, n1, 64);
    k_decode<<<blks((size_t)n1 * 64), TPB, 0, stream>>>(x2bits, x2, (size_t)n1 * 64);
    k_unpool<<<blks((size_t)n1 * 64), TPB, 0, stream>>>(w2, x2, cl2, xu, n1, 64);

    // ---- layer 3: cin=64, cout=32, Ks=1600, Ke=1664 ----------------------
    hipMemsetAsync(S, 0, (size_t)n1 * 1664 * 4, stream);
    k_scatter<<<blks((size_t)E1 * 64), TPB, 0, stream>>>(xu, src1, dst1, ed1k, ed1w, S, 64, 1664, E1);
    k_append<<<blks((size_t)n1 * 64), TPB, 0, stream>>>(xu, S, n1, 64, 1600, 1664);
    k_packw<<<blks((size_t)1664 * 32), TPB, 0, stream>>>(W3, root3, Wext, 1600, 64, 32, 1664);
    {
        int mt = (n1 + 15) / 16;
        dim3 grid((mt + 3) / 4);
        k_wmma_gemm<2><<<grid, gblk, 0, stream>>>(S, Wext, b3, xc, n1, 1664, 32);
    }

    // ---- unpool to level 0 ----------------------------------------------
    k_unpool<<<blks((size_t)n0 * 32), TPB, 0, stream>>>(w1, xc, cl1, x0u, n0, 32);

    // ---- layer 4: cin=32, cout=1, Ks=800, Ke=832 + sigmoid ---------------
    hipMemsetAsync(S, 0, (size_t)n0 * 832 * 4, stream);
    k_scatter<<<blks((size_t)E0 * 32), TPB, 0, stream>>>(x0u, src0, dst0, ed0k, ed0w, S, 32, 832, E0);
    k_append<<<blks((size_t)n0 * 32), TPB, 0, stream>>>(x0u, S, n0, 32, 800, 832);
    k_packw<<<blks((size_t)832 * 1), TPB, 0, stream>>>(W4, root4, Wext, 800, 32, 1, 832);
    {
        dim3 dblk(32, 8);
        dim3 grid((n0 + 7) / 8);
        k_dot_sig<<<grid, dblk, 0, stream>>>(S, Wext, b4, out, n0, 832);
    }
}